// EnhancedCGMNMemory_11845519802577
// MI455X (gfx1250) — compile-verified
//
#include <hip/hip_runtime.h>
#include <hip/hip_bf16.h>
#include <math.h>

// ---------------------------------------------------------------------------
// EnhancedCGMNMemory forward, fused for MI455X (gfx1250, wave32).
// f32 end-to-end; matrix ops via V_WMMA_F32_16X16X4_F32; A tiles staged in LDS.
// ---------------------------------------------------------------------------

typedef __attribute__((ext_vector_type(2))) float v2f;
typedef __attribute__((ext_vector_type(8))) float v8f;

__device__ __forceinline__ v8f wmma4(v2f a, v2f b, v8f c) {
  // D = A(16x4) * B(4x16) + C(16x16), f32.
  // (neg_a, A, neg_b, B, c_mod, C, reuse_a, reuse_b)
  return __builtin_amdgcn_wmma_f32_16x16x4_f32(
      false, a, false, b, (short)0, c, false, false);
}

__device__ __forceinline__ float gelu_exact(float x) {
  return 0.5f * x * (1.0f + erff(x * 0.7071067811865475f));
}

#define NTOK 8192          // B*S
#define T3 48
#define HDIM 256
#define MSLOT 8192
#define KBIG 48
#define KBASE 32
#define ODE_HID 128

// -------------------- kernel A: per-slot ||pos||^2 and curvature weight ----
__global__ __launch_bounds__(256)
void cgmn_slot_prep(const float* __restrict__ pos, const float* __restrict__ curv,
                    const float* __restrict__ alpha_p,
                    float* __restrict__ m2, float* __restrict__ cw) {
  const int m = blockIdx.x * 256 + threadIdx.x;
  const float* p = pos + (size_t)m * T3;
  float s = 0.f;
#pragma unroll
  for (int j = 0; j < T3; ++j) s += p[j] * p[j];
  m2[m] = s;
  const float* c = curv + (size_t)m * 16;
  float n = 0.f;
#pragma unroll
  for (int j = 0; j < 16; ++j) n += c[j] * c[j];
  cw[m] = expf(-alpha_p[0] * sqrtf(n));
}

// -------------------- kernel B: x @ W_proj + b  (8192x1024x48, WMMA) -------
// A tile (16 x 1024, contiguous rows of x) staged once in 64 KB LDS and
// shared by the 3 column-tile waves (3x global-traffic reduction).
__global__ __launch_bounds__(96)
void cgmn_proj(const float* __restrict__ x, const float* __restrict__ Wp,
               const float* __restrict__ bp, float* __restrict__ h) {
  __shared__ float sX[16 * 1024];       // 64 KB (WGP has 320 KB)
  const int t0 = blockIdx.x * 16;
  const int wave = threadIdx.x >> 5;
  const int lane = threadIdx.x & 31;
  const int r = lane & 15;            // A row / B col within tile
  const int khi = (lane >> 4) * 2;    // hi half-wave holds K+2
  const int n0 = wave * 16;           // 3 waves cover the 48 output cols

  {  // contiguous 16K-float block load, vectorized
    const float4* src = (const float4*)(x + (size_t)t0 * 1024);
    float4* dst = (float4*)sX;
    for (int i = threadIdx.x; i < 16 * 256; i += 96) dst[i] = src[i];
  }
  __syncthreads();

  const float* arow = sX + r * 1024 + khi;
  const float* bcol = Wp + (n0 + r);  // B[k][n] = Wp[k*48+n]
  v8f acc = {};
  for (int k = 0; k < 1024; k += 4) {
    v2f a, b;
    a.x = arow[k];     a.y = arow[k + 1];
    b.x = bcol[(size_t)(k + khi) * T3];
    b.y = bcol[(size_t)(k + khi + 1) * T3];
    acc = wmma4(a, b, acc);
  }
  const int n = n0 + r;
  const float bias = bp[n];
#pragma unroll
  for (int v = 0; v < 8; ++v) {
    const int m = v + ((lane >> 4) << 3);
    h[(size_t)(t0 + m) * T3 + n] = acc[v] + bias;
  }
}

// -------------------- kernel B2: LN(48) + GELU + 2 ODE steps + ||q||^2 -----
__global__ __launch_bounds__(256)
void cgmn_ln_ode(const float* __restrict__ g1, const float* __restrict__ b1ln,
                 const float* __restrict__ W1, const float* __restrict__ b1,
                 const float* __restrict__ W2, const float* __restrict__ b2,
                 float* __restrict__ h, float* __restrict__ q2) {
  __shared__ float sW1[T3 * ODE_HID];
  __shared__ float sW2[ODE_HID * T3];
  __shared__ float sb1[ODE_HID];
  __shared__ float sb2[T3];
  for (int i = threadIdx.x; i < T3 * ODE_HID; i += 256) sW1[i] = W1[i];
  for (int i = threadIdx.x; i < ODE_HID * T3; i += 256) sW2[i] = W2[i];
  for (int i = threadIdx.x; i < ODE_HID; i += 256) sb1[i] = b1[i];
  for (int i = threadIdx.x; i < T3; i += 256) sb2[i] = b2[i];
  __syncthreads();

  const int t = blockIdx.x * 256 + threadIdx.x;
  float* row = h + (size_t)t * T3;
  float hh[T3];
  float mu = 0.f;
#pragma unroll
  for (int d = 0; d < T3; ++d) { hh[d] = row[d]; mu += hh[d]; }
  mu *= (1.f / T3);
  float var = 0.f;
#pragma unroll
  for (int d = 0; d < T3; ++d) { const float c = hh[d] - mu; var += c * c; }
  const float rs = rsqrtf(var * (1.f / T3) + 1e-5f);
#pragma unroll
  for (int d = 0; d < T3; ++d)
    hh[d] = gelu_exact((hh[d] - mu) * rs * g1[d] + b1ln[d]);

  for (int step = 0; step < 2; ++step) {
    float acc[T3];
#pragma unroll
    for (int j = 0; j < T3; ++j) acc[j] = sb2[j];
#pragma unroll 1
    for (int i = 0; i < ODE_HID; ++i) {
      float z = sb1[i];
#pragma unroll
      for (int d = 0; d < T3; ++d) z += hh[d] * sW1[d * ODE_HID + i];
      z = tanhf(z);
#pragma unroll
      for (int j = 0; j < T3; ++j) acc[j] += z * sW2[i * T3 + j];
    }
#pragma unroll
    for (int d = 0; d < T3; ++d) hh[d] += 0.5f * acc[d];
  }
  float s2 = 0.f;
#pragma unroll
  for (int d = 0; d < T3; ++d) { row[d] = hh[d]; s2 += hh[d] * hh[d]; }
  q2[t] = s2;
}

// ---- kernel C: fused dist GEMM (WMMA, K=48) + filtered per-token top-48 ---
// 8 waves, 128-slot chunks (64 barrier iterations instead of 128).
__global__ __launch_bounds__(256)
void cgmn_dist_topk(const float* __restrict__ h, const float* __restrict__ pos,
                    const float* __restrict__ q2, const float* __restrict__ m2,
                    const float* __restrict__ cw,
                    float* __restrict__ topv, int* __restrict__ topi) {
  __shared__ float tv[16][KBIG];
  __shared__ int   ti[16][KBIG];
  __shared__ float qv[16][128];
  __shared__ int   qi[16][128];
  __shared__ int   qn[16];

  const int t0 = blockIdx.x * 16;
  const int wave = threadIdx.x >> 5;
  const int lane = threadIdx.x & 31;
  const int r = lane & 15;
  const int khi = (lane >> 4) * 2;

  for (int i = threadIdx.x; i < 16 * KBIG; i += 256) {
    tv[i / KBIG][i % KBIG] = 1e30f;
    ti[i / KBIG][i % KBIG] = 0;
  }
  __syncthreads();

  const float* arow = h + (size_t)(t0 + r) * T3 + khi;
  float ar[12][2];                       // hoist invariant A tile into regs
#pragma unroll
  for (int k = 0; k < 12; ++k) { ar[k][0] = arow[k * 4]; ar[k][1] = arow[k * 4 + 1]; }
  float q2r[8];
#pragma unroll
  for (int v = 0; v < 8; ++v) q2r[v] = q2[t0 + v + ((lane >> 4) << 3)];

  for (int c = 0; c < MSLOT; c += 128) {
    const int n0 = c + wave * 16;
    const float* brow = pos + (size_t)(n0 + r) * T3 + khi;
    if (c + 128 < MSLOT)
      __builtin_prefetch(pos + (size_t)(n0 + 128 + r) * T3, 0, 1);
    v8f acc = {};
#pragma unroll
    for (int k = 0; k < 12; ++k) {
      v2f a, b;
      a.x = ar[k][0]; a.y = ar[k][1];
      b.x = brow[k * 4]; b.y = brow[k * 4 + 1];
      acc = wmma4(a, b, acc);
    }
    if (threadIdx.x < 16) qn[threadIdx.x] = 0;
    __syncthreads();
    const int gn = n0 + r;
    const float m2v = m2[gn], cwv = cw[gn];
#pragma unroll
    for (int v = 0; v < 8; ++v) {
      const int m = v + ((lane >> 4) << 3);
      float d = fmaxf(q2r[v] + m2v - 2.f * acc[v], 0.f) * cwv;
      if (d < tv[m][KBIG - 1]) {            // conservative filter
        const int p = atomicAdd(&qn[m], 1); // LDS atomic (ds_add_rtn_u32)
        qv[m][p] = d;
        qi[m][p] = gn;
      }
    }
    __syncthreads();
    if (threadIdx.x < 16) {
      const int t = threadIdx.x;
      const int ncand = qn[t];
#pragma unroll 1
      for (int e = 0; e < ncand; ++e) {
        const float v = qv[t][e];
        if (v < tv[t][KBIG - 1]) {
          int p = KBIG - 1;
          while (p > 0 && tv[t][p - 1] > v) {
            tv[t][p] = tv[t][p - 1]; ti[t][p] = ti[t][p - 1]; --p;
          }
          tv[t][p] = v; ti[t][p] = qi[t][e];
        }
      }
    }
    __syncthreads();
  }
  if (threadIdx.x < 16) {
    const int t = threadIdx.x;
    for (int j = 0; j < KBIG; ++j) {
      topv[(size_t)(t0 + t) * KBIG + j] = tv[t][j];
      topi[(size_t)(t0 + t) * KBIG + j] = ti[t][j];
    }
  }
}

// -------------------- kernel D: deterministic mean(top1) -> keep count -----
__global__ __launch_bounds__(256)
void cgmn_fire(const float* __restrict__ topv, int* __restrict__ keep) {
  __shared__ float red[256];
  float s = 0.f;
  for (int i = threadIdx.x; i < NTOK; i += 256) s += topv[(size_t)i * KBIG];
  red[threadIdx.x] = s;
  __syncthreads();
  for (int off = 128; off > 0; off >>= 1) {
    if (threadIdx.x < off) red[threadIdx.x] += red[threadIdx.x + off];
    __syncthreads();
  }
  if (threadIdx.x == 0) {
    const float mean = red[0] * (1.f / NTOK);
    keep[0] = (mean < 0.7f * 1.0f) ? KBIG : KBASE;  // LB_DROP_RATIO*LB_TOP1_AVG
  }
}

// -------------------- kernel E: softmax + L2-resident gather + weighted sum -
__global__ __launch_bounds__(256)
void cgmn_attend(const float* __restrict__ topv, const int* __restrict__ topi,
                 const int* __restrict__ keepp, const float* __restrict__ mem,
                 float* __restrict__ att) {
  __shared__ float wv[8][KBIG];
  __shared__ int   wi[8][KBIG];
  __shared__ float winv[8];
  const int wave = threadIdx.x >> 5;
  const int lane = threadIdx.x & 31;
  const int tok = blockIdx.x * 8 + wave;
  const int keep = keepp[0];
  const float v0 = topv[(size_t)tok * KBIG];
  for (int j = lane; j < KBIG; j += 32) {
    const float v = topv[(size_t)tok * KBIG + j];
    wv[wave][j] = (j < keep) ? expf(v0 - v) : 0.f;  // softmax(-d), max = -v0
    wi[wave][j] = topi[(size_t)tok * KBIG + j];
  }
  __syncthreads();
  if (lane == 0) {
    float s = 0.f;
    for (int j = 0; j < KBIG; ++j) s += wv[wave][j];
    winv[wave] = 1.f / s;
  }
  __syncthreads();
  const float inv = winv[wave];
  float4 a0 = {0, 0, 0, 0}, a1 = {0, 0, 0, 0};
#pragma unroll 1
  for (int j = 0; j < keep; ++j) {
    const float w = wv[wave][j] * inv;
    const float4* p = (const float4*)(mem + (size_t)wi[wave][j] * HDIM) + lane * 2;
    const float4 m0 = p[0], m1 = p[1];
    a0.x += w * m0.x; a0.y += w * m0.y; a0.z += w * m0.z; a0.w += w * m0.w;
    a1.x += w * m1.x; a1.y += w * m1.y; a1.z += w * m1.z; a1.w += w * m1.w;
  }
  float4* o = (float4*)(att + (size_t)tok * HDIM) + lane * 2;
  o[0] = a0; o[1] = a1;
}

// -------------------- kernel F1: attended @ W_out + b  (8192x256x1024, WMMA)
// A tile (16 x 256) staged in 16 KB LDS: shared by all 8 waves and all 8
// column sweeps (64x reduction of A global traffic vs direct loads).
__global__ __launch_bounds__(256)
void cgmn_outgemm(const float* __restrict__ att, const float* __restrict__ Wo,
                  const float* __restrict__ bo, float* __restrict__ out) {
  __shared__ float sA[16 * HDIM];       // 16 KB
  const int t0 = blockIdx.x * 16;
  const int wave = threadIdx.x >> 5;
  const int lane = threadIdx.x & 31;
  const int r = lane & 15;
  const int khi = (lane >> 4) * 2;

  {  // contiguous 4K-float block load, vectorized (4 float4 per thread)
    const float4* src = (const float4*)(att + (size_t)t0 * HDIM);
    float4* dst = (float4*)sA;
#pragma unroll
    for (int i = 0; i < 4; ++i) dst[threadIdx.x + 256 * i] = src[threadIdx.x + 256 * i];
  }
  __syncthreads();

  const float* arow = sA + r * HDIM + khi;
  for (int s = 0; s < 8; ++s) {
    const int n0 = (s * 8 + wave) * 16;
    const int n = n0 + r;
    const float bias = bo[n];
    v8f acc;
#pragma unroll
    for (int v = 0; v < 8; ++v) acc[v] = bias;   // C init = bias (same col)
    for (int k = 0; k < HDIM; k += 4) {
      v2f a, b;
      a.x = arow[k]; a.y = arow[k + 1];
      b.x = Wo[(size_t)(k + khi) * 1024 + n];
      b.y = Wo[(size_t)(k + khi + 1) * 1024 + n];
      acc = wmma4(a, b, acc);
    }
#pragma unroll
    for (int v = 0; v < 8; ++v) {
      const int m = v + ((lane >> 4) << 3);
      out[(size_t)(t0 + m) * 1024 + n] = acc[v];
    }
  }
}

// -------------------- kernel F2: in-place LN(1024) + GELU ------------------
__global__ __launch_bounds__(256)
void cgmn_ln_gelu(const float* __restrict__ g, const float* __restrict__ b,
                  float* __restrict__ out) {
  __shared__ float red[256];
  __shared__ float s_mu, s_rs;
  float4* row = (float4*)(out + (size_t)blockIdx.x * 1024);
  const float4 x = row[threadIdx.x];
  red[threadIdx.x] = x.x + x.y + x.z + x.w;
  __syncthreads();
  for (int off = 128; off > 0; off >>= 1) {
    if (threadIdx.x < off) red[threadIdx.x] += red[threadIdx.x + off];
    __syncthreads();
  }
  if (threadIdx.x == 0) s_mu = red[0] * (1.f / 1024.f);
  __syncthreads();
  const float mu = s_mu;
  const float c0 = x.x - mu, c1 = x.y - mu, c2 = x.z - mu, c3 = x.w - mu;
  red[threadIdx.x] = c0 * c0 + c1 * c1 + c2 * c2 + c3 * c3;
  __syncthreads();
  for (int off = 128; off > 0; off >>= 1) {
    if (threadIdx.x < off) red[threadIdx.x] += red[threadIdx.x + off];
    __syncthreads();
  }
  if (threadIdx.x == 0) s_rs = rsqrtf(red[0] * (1.f / 1024.f) + 1e-5f);
  __syncthreads();
  const float rs = s_rs;
  const int n = threadIdx.x * 4;
  float4 y;
  y.x = gelu_exact(c0 * rs * g[n + 0] + b[n + 0]);
  y.y = gelu_exact(c1 * rs * g[n + 1] + b[n + 1]);
  y.z = gelu_exact(c2 * rs * g[n + 2] + b[n + 2]);
  y.w = gelu_exact(c3 * rs * g[n + 3] + b[n + 3]);
  row[threadIdx.x] = y;
}

// ---------------------------------------------------------------------------
extern "C" void kernel_launch(void* const* d_in, const int* in_sizes, int n_in,
                              void* d_out, int out_size, void* d_ws, size_t ws_size,
                              hipStream_t stream) {
  (void)in_sizes; (void)n_in; (void)out_size; (void)ws_size;
  const float* x     = (const float*)d_in[0];
  const float* Wp    = (const float*)d_in[1];
  const float* bp    = (const float*)d_in[2];
  const float* ln1g  = (const float*)d_in[3];
  const float* ln1b  = (const float*)d_in[4];
  const float* oW1   = (const float*)d_in[5];
  const float* ob1   = (const float*)d_in[6];
  const float* oW2   = (const float*)d_in[7];
  const float* ob2   = (const float*)d_in[8];
  const float* mem   = (const float*)d_in[9];
  const float* pos   = (const float*)d_in[10];
  const float* curv  = (const float*)d_in[11];
  const float* alpha = (const float*)d_in[12];
  const float* Wo    = (const float*)d_in[13];
  const float* bo    = (const float*)d_in[14];
  const float* ln2g  = (const float*)d_in[15];
  const float* ln2b  = (const float*)d_in[16];
  float* out = (float*)d_out;

  // workspace layout (floats): ~12.6 MB total
  float* ws   = (float*)d_ws;
  float* h    = ws;                       // 8192*48
  float* q2   = ws + 393216;              // 8192
  float* m2   = ws + 401408;              // 8192
  float* cw   = ws + 409600;              // 8192
  float* topv = ws + 417792;              // 8192*48
  int*   topi = (int*)(ws + 811008);      // 8192*48
  float* att  = ws + 1204224;             // 8192*256
  int*   keep = (int*)(ws + 3301376);     // 1

  cgmn_slot_prep<<<MSLOT / 256, 256, 0, stream>>>(pos, curv, alpha, m2, cw);
  cgmn_proj<<<NTOK / 16, 96, 0, stream>>>(x, Wp, bp, h);
  cgmn_ln_ode<<<NTOK / 256, 256, 0, stream>>>(ln1g, ln1b, oW1, ob1, oW2, ob2, h, q2);
  cgmn_dist_topk<<<NTOK / 16, 256, 0, stream>>>(h, pos, q2, m2, cw, topv, topi);
  cgmn_fire<<<1, 256, 0, stream>>>(topv, keep);
  cgmn_attend<<<NTOK / 8, 256, 0, stream>>>(topv, topi, keep, mem, att);
  cgmn_outgemm<<<NTOK / 16, 256, 0, stream>>>(att, Wo, bo, out);
  cgmn_ln_gelu<<<NTOK, 256, 0, stream>>>(ln2g, ln2b, out);
}